// TensorBasis_15599321219725
// MI455X (gfx1250) — compile-verified
//
#include <hip/hip_runtime.h>

typedef float v2f __attribute__((ext_vector_type(2)));
typedef float v8f __attribute__((ext_vector_type(8)));

// out[i,p] = sum_{m,n} A[i,m]*B[i,n]*C[m,n,p]
//          = sum_k P[i,k]*W[k,p],  k = m*5+n (25 values, padded to 28),
// W[k,p] = C_flat[k*5+p] (p<5, padded to 16 columns with zeros).
// One 16-edge tile per wave iteration; 7x V_WMMA_F32_16X16X4_F32 chained on acc.
__global__ __launch_bounds__(256) void cg_combine_wmma(
    const float* __restrict__ A, const float* __restrict__ B,
    const float* __restrict__ C, float* __restrict__ out,
    int nEdges, int nTiles)
{
    const int lane = threadIdx.x & 31;   // wave32
    const int l    = lane & 15;          // position within 16-wide half
    const int hi   = lane >> 4;          // 0: K pair (4j,4j+1), 1: (4j+2,4j+3)

    // ---- Constant B-operand W (28x16, zero padded), held in 14 VGPRs.
    // B-matrix layout mirrors A: N = lane%16; VGPR0 holds K=4j+2*hi, VGPR1 K+1.
    v2f W[7];
#pragma unroll
    for (int j = 0; j < 7; ++j) {
        const int k0 = 4 * j + 2 * hi;
        const int k1 = k0 + 1;
        float w0 = (k0 < 25 && l < 5) ? C[k0 * 5 + l] : 0.0f;
        float w1 = (k1 < 25 && l < 5) ? C[k1 * 5 + l] : 0.0f;
        v2f w; w.x = w0; w.y = w1;
        W[j] = w;
    }

    const int wavesPerBlock = blockDim.x >> 5;
    const int waveId   = blockIdx.x * wavesPerBlock + (threadIdx.x >> 5);
    const int numWaves = gridDim.x * wavesPerBlock;
    const long strideEdges = (long)numWaves * 16;   // grid-stride in edges

    for (int tile = waveId; tile < nTiles; tile += numWaves) {
        const int  edge     = tile * 16 + l;        // lanes L and L+16 share edge L
        const bool fullTile = (tile * 16 + 16) <= nEdges;   // wave-uniform

        float a[5], b[5];
        if (fullTile) {
            const long base = (long)edge * 5;
#pragma unroll
            for (int q = 0; q < 5; ++q) { a[q] = A[base + q]; b[q] = B[base + q]; }
            // Prefetch next grid-stride tile's rows (speculative, no counter cost).
            __builtin_prefetch(&A[base + strideEdges * 5], 0, 1);
            __builtin_prefetch(&B[base + strideEdges * 5], 0, 1);
        } else if (edge < nEdges) {
            const long base = (long)edge * 5;
#pragma unroll
            for (int q = 0; q < 5; ++q) { a[q] = A[base + q]; b[q] = B[base + q]; }
        } else {
#pragma unroll
            for (int q = 0; q < 5; ++q) { a[q] = 0.0f; b[q] = 0.0f; }
        }

        v8f acc = {};
#pragma unroll
        for (int j = 0; j < 7; ++j) {
            // Outer-product entries at compile-time indices; select per half-wave.
            const int kA = 4 * j, kB = 4 * j + 1, kC = 4 * j + 2, kD = 4 * j + 3;
            const float pA = (kA < 25) ? a[kA / 5] * b[kA % 5] : 0.0f;
            const float pB = (kB < 25) ? a[kB / 5] * b[kB % 5] : 0.0f;
            const float pC = (kC < 25) ? a[kC / 5] * b[kC % 5] : 0.0f;
            const float pD = (kD < 25) ? a[kD / 5] * b[kD % 5] : 0.0f;
            v2f P;
            P.x = hi ? pC : pA;
            P.y = hi ? pD : pB;
            // D = A(16x4) * B(4x16) + C ; 8 args: neg_a, A, neg_b, B, c_mod, C, reuse_a, reuse_b
            acc = __builtin_amdgcn_wmma_f32_16x16x4_f32(
                false, P, false, W[j], (short)0, acc, false, false);
        }

        // D layout: VGPR r -> row (tile*16 + r + 8*hi), column p = lane%16 (p<5 valid).
        if (fullTile) {
            if (l < 5) {   // single exec-mask; 8 NT stores at immediate offsets
                float* orow = out + (long)(tile * 16 + 8 * hi) * 5 + l;
#pragma unroll
                for (int r = 0; r < 8; ++r)
                    __builtin_nontemporal_store(acc[r], orow + r * 5);
            }
        } else {
#pragma unroll
            for (int r = 0; r < 8; ++r) {
                const int row = tile * 16 + r + 8 * hi;
                if (l < 5 && row < nEdges) out[(long)row * 5 + l] = acc[r];
            }
        }
    }
}

extern "C" void kernel_launch(void* const* d_in, const int* in_sizes, int n_in,
                              void* d_out, int out_size, void* d_ws, size_t ws_size,
                              hipStream_t stream) {
    const float* A = (const float*)d_in[0];
    const float* B = (const float*)d_in[1];
    const float* C = (const float*)d_in[2];
    float* out = (float*)d_out;

    const int nEdges = in_sizes[0] / 5;
    const int nTiles = (nEdges + 15) / 16;

    const int block = 256;                       // 8 wave32 per block
    int blocks = (nTiles + 7) / 8;
    if (blocks > 32768) blocks = 32768;          // grid-stride amortizes W setup
    if (blocks < 1) blocks = 1;

    cg_combine_wmma<<<blocks, block, 0, stream>>>(A, B, C, out, nEdges, nTiles);
}